// CrossCompressUnit_83279415870031
// MI455X (gfx1250) — compile-verified
//
#include <hip/hip_runtime.h>

typedef __attribute__((ext_vector_type(2))) float v2f;
typedef __attribute__((ext_vector_type(8))) float v8f;

#define DIM 128
#define K_SEGS 4                        // K split across 4 waves per 16-row tile
#define SEG_STEPS (DIM / 4 / K_SEGS)    // 8 wmma steps of K=4 per wave
#define TILES_PER_BLOCK 2
#define ROWS_PER_BLOCK (TILES_PER_BLOCK * 16)   // 32 rows / block
#define D1_THREADS 256                  // 8 waves = 2 tiles x 4 K-segments
#define D2_THREADS 256

// Algebraic collapse of the reference:
//   v_out[b,i] = v[b,i]*(e[b].w_vv) + e[b,i]*(v[b].w_ev) + b_vv
//   e_out[b,i] = v[b,i]*(e[b].w_ve) + e[b,i]*(v[b].w_ee) + b_ee
// Kernel 1: per-row dot4 via fp32 WMMA (16x16x4), K-split 4-ways, LDS reduce.
__global__ __launch_bounds__(D1_THREADS)
void cross_dots_kernel(const float* __restrict__ v,
                       const float* __restrict__ e,
                       const float* __restrict__ w_vv,
                       const float* __restrict__ w_ev,
                       const float* __restrict__ w_ve,
                       const float* __restrict__ w_ee,
                       float* __restrict__ dots /* [B][4] */)
{
    __shared__ float s_part[K_SEGS][4][ROWS_PER_BLOCK];   // [seg][dot][rowInBlock]

    const int tid     = threadIdx.x;
    const int wave    = tid >> 5;
    const int lane    = tid & 31;
    const int laneLow = lane & 15;       // M (A row) or N (B/D column)
    const int half    = lane >> 4;       // K-half selector for A/B fragments

    const int tileInBlock = wave >> 2;   // 0..1
    const int seg         = wave & 3;    // K segment 0..3

    const int rowBaseBlock = blockIdx.x * ROWS_PER_BLOCK;
    const int rowBaseTile  = rowBaseBlock + tileInBlock * 16;
    const int myRow        = rowBaseTile + laneLow;

    const float* vRow = v + (size_t)myRow * DIM;
    const float* eRow = e + (size_t)myRow * DIM;

    // B columns: N==0,1 carry weight vectors, rest zero. Pointers valid for all
    // lanes; mask applied to loaded VALUE (v_cndmask, no EXEC-predicated loads).
    const bool   bActive = (laneLow < 2);
    const float* wForV   = (laneLow == 0) ? w_ev : w_ee;
    const float* wForE   = (laneLow == 0) ? w_vv : w_ve;

    v8f accV = {};   // col0: v.w_ev   col1: v.w_ee
    v8f accE = {};   // col0: e.w_vv   col1: e.w_ve

    const int kBase = seg * (DIM / K_SEGS);               // 32 floats per segment

    #pragma unroll
    for (int k = 0; k < SEG_STEPS; ++k) {
        const int colA = kBase + k * 4 + half * 2;        // f32 A layout: V0:K0|K2, V1:K1|K3
        v2f aV = *(const v2f*)(vRow + colA);
        v2f aE = *(const v2f*)(eRow + colA);
        v2f tV = *(const v2f*)(wForV + colA);
        v2f tE = *(const v2f*)(wForE + colA);
        v2f bz = {0.0f, 0.0f};
        v2f bV = bActive ? tV : bz;
        v2f bE = bActive ? tE : bz;
        accV = __builtin_amdgcn_wmma_f32_16x16x4_f32(false, aV, false, bV,
                                                     (short)0, accV, false, false);
        accE = __builtin_amdgcn_wmma_f32_16x16x4_f32(false, aE, false, bE,
                                                     (short)0, accE, false, false);
    }

    // D layout (16x16 f32): lanes n / n+16 hold column n; M = vgpr idx (+8 upper half).
    // dot0 = e.w_vv (accE col0), dot1 = v.w_ev (accV col0),
    // dot2 = e.w_ve (accE col1), dot3 = v.w_ee (accV col1)
    if (laneLow < 2) {
        const int base = tileInBlock * 16 + half * 8;
        const int dE = (laneLow == 0) ? 0 : 2;
        const int dV = (laneLow == 0) ? 1 : 3;
        #pragma unroll
        for (int r = 0; r < 8; ++r) {
            s_part[seg][dE][base + r] = accE[r];
            s_part[seg][dV][base + r] = accV[r];
        }
    }
    __syncthreads();

    // Reduce the 4 K-segments; 128 (row,dot) sums per block.
    if (tid < ROWS_PER_BLOCK * 4) {
        const int row = tid & (ROWS_PER_BLOCK - 1);
        const int dot = tid >> 5;
        const float s = s_part[0][dot][row] + s_part[1][dot][row]
                      + s_part[2][dot][row] + s_part[3][dot][row];
        dots[(size_t)(rowBaseBlock + row) * 4 + dot] = s;
    }
}

// Kernel 2: streaming axpy, exactly one float4 per thread (max concurrency).
__global__ __launch_bounds__(D2_THREADS)
void cross_axpy_kernel(const float* __restrict__ v,
                       const float* __restrict__ e,
                       const float4* __restrict__ dots,   // [B]: {e.wvv, v.wev, e.wve, v.wee}
                       const float* __restrict__ b_vv,
                       const float* __restrict__ b_ee,
                       float* __restrict__ v_out,
                       float* __restrict__ e_out)
{
    const int g   = blockIdx.x * D2_THREADS + threadIdx.x;   // float4 index
    const int row = g >> 5;                                  // DIM/4 = 32 quads per row

    const float4 vv = ((const float4*)v)[g];
    const float4 ee = ((const float4*)e)[g];
    const float4 d  = dots[row];            // wave-uniform (one row per wave)
    const float bvv = b_vv[0];
    const float bee = b_ee[0];

    float4 vo, eo;
    vo.x = vv.x * d.x + ee.x * d.y + bvv;
    vo.y = vv.y * d.x + ee.y * d.y + bvv;
    vo.z = vv.z * d.x + ee.z * d.y + bvv;
    vo.w = vv.w * d.x + ee.w * d.y + bvv;
    eo.x = vv.x * d.z + ee.x * d.w + bee;
    eo.y = vv.y * d.z + ee.y * d.w + bee;
    eo.z = vv.z * d.z + ee.z * d.w + bee;
    eo.w = vv.w * d.z + ee.w * d.w + bee;
    ((float4*)v_out)[g] = vo;
    ((float4*)e_out)[g] = eo;
}

extern "C" void kernel_launch(void* const* d_in, const int* in_sizes, int n_in,
                              void* d_out, int out_size, void* d_ws, size_t ws_size,
                              hipStream_t stream) {
    const float* v    = (const float*)d_in[0];
    const float* e    = (const float*)d_in[1];
    const float* w_vv = (const float*)d_in[2];
    const float* b_vv = (const float*)d_in[3];
    const float* w_ev = (const float*)d_in[4];
    const float* w_ve = (const float*)d_in[5];
    const float* w_ee = (const float*)d_in[6];
    const float* b_ee = (const float*)d_in[7];

    const int Btot = in_sizes[0] / DIM;                 // 8192
    float* v_out = (float*)d_out;
    float* e_out = (float*)d_out + (size_t)Btot * DIM;  // tuple concat order
    float* dots  = (float*)d_ws;                        // B*4 floats = 128 KB

    // Phase 1: 256 blocks x 8 waves = 2048 waves computing per-row dot4 via WMMA.
    dim3 g1(Btot / ROWS_PER_BLOCK);
    cross_dots_kernel<<<g1, D1_THREADS, 0, stream>>>(v, e, w_vv, w_ev, w_ve, w_ee, dots);

    // Phase 2: one float4 per thread -> 1024 blocks x 256 threads = 8192 waves.
    dim3 g2((Btot * (DIM / 4)) / D2_THREADS);
    cross_axpy_kernel<<<g2, D2_THREADS, 0, stream>>>(v, e, (const float4*)dots,
                                                     b_vv, b_ee, v_out, e_out);
}